// MultiHeadLatentAttention_38336878084511
// MI455X (gfx1250) — compile-verified
//
#include <hip/hip_runtime.h>
#include <hip/hip_bf16.h>

typedef float v2f __attribute__((ext_vector_type(2)));
typedef float v8f __attribute__((ext_vector_type(8)));

#define T_LEN 2048
#define HDIM  64

// D = A(16x4) * B(4x16) + C  in fp32
static __device__ __forceinline__ v8f wmma4(v2f a, v2f b, v8f c) {
  return __builtin_amdgcn_wmma_f32_16x16x4_f32(false, a, false, b, (short)0, c, false, false);
}

#define SHFL16(x) __shfl_xor((x), 16, 32)

// ---------------------------------------------------------------------------
// Kernel 1: combine latent projections.
//   Wq  (64x64)  = W_UQ (64x512)  @ W_DQ (512x64)
//   Wkv (128x64) = W_UKV(128x1024)@ W_DKV(1024x64)
// ---------------------------------------------------------------------------
__global__ void __launch_bounds__(256) combine_w_kernel(
    const float* __restrict__ W_UQ, const float* __restrict__ W_DQ,
    const float* __restrict__ W_UKV, const float* __restrict__ W_DKV,
    float* __restrict__ Wq, float* __restrict__ Wkv) {
  const int e = blockIdx.x * 256 + threadIdx.x;     // 12288 entries
  if (e < 4096) {
    const int dp = e >> 6, d = e & 63;
    float s = 0.f;
    for (int l = 0; l < 512; ++l)
      s = fmaf(W_UQ[dp * 512 + l], W_DQ[l * 64 + d], s);
    Wq[e] = s;
  } else {
    const int e2 = e - 4096;
    const int i = e2 >> 6, d = e2 & 63;
    float s = 0.f;
    for (int l = 0; l < 1024; ++l)
      s = fmaf(W_UKV[i * 1024 + l], W_DKV[l * 64 + d], s);
    Wkv[e2] = s;
  }
}

// ---------------------------------------------------------------------------
// Kernel 2: fused QKV + RMSNorm. One thread per (b,t,h).
//   Q[bh][t][128] = concat(rmsnorm(x@Wq^T)*qw,  qn@W_QR^T)
//   K[bh][t][128] = concat(rmsnorm(x@Wkv[0:64]^T)*kw,  x@W_KR^T)
//   V[bh][t][64]  = x@Wkv[64:128]^T
// ---------------------------------------------------------------------------
static __device__ __forceinline__ float dot64(const float* __restrict__ w,
                                              const float (&xv)[64]) {
  float s0 = 0.f, s1 = 0.f, s2 = 0.f, s3 = 0.f;
#pragma unroll
  for (int i = 0; i < 16; ++i) {
    float4 wv = *(const float4*)(w + 4 * i);
    s0 = fmaf(wv.x, xv[4 * i + 0], s0);
    s1 = fmaf(wv.y, xv[4 * i + 1], s1);
    s2 = fmaf(wv.z, xv[4 * i + 2], s2);
    s3 = fmaf(wv.w, xv[4 * i + 3], s3);
  }
  return (s0 + s1) + (s2 + s3);
}

__global__ void __launch_bounds__(256) mla_qkv_kernel(
    const float* __restrict__ x, const float* __restrict__ Wq,
    const float* __restrict__ Wkv, const float* __restrict__ W_QR,
    const float* __restrict__ W_KR, const float* __restrict__ qnw,
    const float* __restrict__ knw, float* __restrict__ Qo,
    float* __restrict__ Ko, float* __restrict__ Vo) {
  const int gid = blockIdx.x * 256 + threadIdx.x;   // B*T*H = 65536
  const int h = gid & 15;
  const int t = (gid >> 4) & (T_LEN - 1);
  const int b = gid >> 15;
  const int bh = b * 16 + h;

  const float* xp = x + ((size_t)(b * T_LEN + t)) * 1024 + h * HDIM;
  float xv[64];
#pragma unroll
  for (int i = 0; i < 16; ++i) {
    float4 v = *(const float4*)(xp + 4 * i);
    xv[4 * i + 0] = v.x; xv[4 * i + 1] = v.y;
    xv[4 * i + 2] = v.z; xv[4 * i + 3] = v.w;
  }

  float* qrow = Qo + ((size_t)bh * T_LEN + t) * 128;
  float* krow = Ko + ((size_t)bh * T_LEN + t) * 128;
  float* vrow = Vo + ((size_t)bh * T_LEN + t) * HDIM;

  // k_r (raw x)
  for (int j = 0; j < 64; ++j) krow[64 + j] = dot64(W_KR + j * 64, xv);
  // v
  for (int j = 0; j < 64; ++j) vrow[j] = dot64(Wkv + (64 + j) * 64, xv);
  // k_c + rmsnorm
  float msk = 0.f;
  for (int j = 0; j < 64; ++j) {
    float v = dot64(Wkv + j * 64, xv);
    krow[j] = v;
    msk = fmaf(v, v, msk);
  }
  const float rk = rsqrtf(msk * (1.0f / 64.0f) + 1.1920929e-07f);
#pragma unroll
  for (int i = 0; i < 64; ++i) krow[i] = krow[i] * rk * knw[i];
  // q_c + rmsnorm
  float msq = 0.f;
  for (int j = 0; j < 64; ++j) {
    float v = dot64(Wq + j * 64, xv);
    qrow[j] = v;
    msq = fmaf(v, v, msq);
  }
  const float rq = rsqrtf(msq * (1.0f / 64.0f) + 1.1920929e-07f);
  float qn[64];
#pragma unroll
  for (int i = 0; i < 64; ++i) {
    float v = qrow[i] * rq * qnw[i];
    qn[i] = v;
    qrow[i] = v;
  }
  // q_r (from normalized q_c)
  for (int j = 0; j < 64; ++j) qrow[64 + j] = dot64(W_QR + j * 64, qn);
}

// ---------------------------------------------------------------------------
// Kernel 3: causal flash attention + W_O epilogue.
// One wave32 per 16-query tile; transposed formulation:
//   S^T = K_tile(16x128) x Q^T   -> softmax stats per-lane (query = lane)
//   O^T = V^T x P^T              -> 64x16 accumulator, 4 M-tiles of v8f
//   out^T = W_O x (O^T / l)      -> WMMA epilogue
// P^T C-layout -> B-operand fragments via __shfl_xor(...,16) half swaps.
// ---------------------------------------------------------------------------
__global__ void __launch_bounds__(128) mla_attn_kernel(
    const float* __restrict__ Qm, const float* __restrict__ Km,
    const float* __restrict__ Vm, const float* __restrict__ W_O,
    float* __restrict__ out) {
  const int lane = threadIdx.x & 31;
  const int wid = threadIdx.x >> 5;
  const int n = lane & 15;       // query column / A-matrix row
  const int hi = lane >> 4;      // K-dim half select
  const int bh = blockIdx.y;     // 32 (b,h) pairs
  const int qt = blockIdx.x * 4 + wid;   // 128 query tiles
  const int q0 = qt << 4;

  const size_t qkBase = (size_t)bh * T_LEN * 128;

  // Q fragments: serve as B operand (Q^T) for every key tile. 64 VGPRs.
  const float* qrow = Qm + qkBase + (size_t)(q0 + n) * 128 + 2 * hi;
  v2f qf[32];
#pragma unroll
  for (int c = 0; c < 32; ++c) qf[c] = *(const v2f*)(qrow + 4 * c);

  v8f o[4] = {};                 // O^T accumulator (64 v-dims x 16 queries)
  float mrow = -__builtin_inff();
  float lrow = 0.f;

  for (int kt = 0; kt <= qt; ++kt) {
    const int k0 = kt << 4;
    const float* krow = Km + qkBase + (size_t)(k0 + n) * 128 + 2 * hi;

    // S^T = K x Q^T over d=128 (32 chunks of 4)
    v8f s = {};
#pragma unroll
    for (int c = 0; c < 32; ++c) {
      v2f kf = *(const v2f*)(krow + 4 * c);
      s = wmma4(kf, qf[c], s);
    }

    const bool diag = (kt == qt);
    float p[8];
    float tmax = -__builtin_inff();
#pragma unroll
    for (int r = 0; r < 8; ++r) {
      float v = s[r] * 0.125f;                    // HD^-0.5
      if (diag && ((r + 8 * hi) > n)) v = -__builtin_inff();  // key > query
      p[r] = v;
      tmax = fmaxf(tmax, v);
    }
    tmax = fmaxf(tmax, SHFL16(tmax));             // keys 0-7 vs 8-15
    const float mnew = fmaxf(mrow, tmax);
    const float alpha = __expf(mrow - mnew);
    float psum = 0.f;
#pragma unroll
    for (int r = 0; r < 8; ++r) {
      float e = __expf(p[r] - mnew);
      p[r] = e;
      psum += e;
    }
    psum += SHFL16(psum);
    lrow = lrow * alpha + psum;
    mrow = mnew;
#pragma unroll
    for (int mt = 0; mt < 4; ++mt)
#pragma unroll
      for (int r = 0; r < 8; ++r) o[mt][r] *= alpha;

    // O^T += V^T x P^T  (keys in chunks of 4)
    const float* vbase = Vm + (size_t)bh * T_LEN * HDIM + (size_t)k0 * HDIM + n;
#pragma unroll
    for (int kc = 0; kc < 4; ++kc) {
      float b0, b1;
      if (kc == 0) { float y0 = SHFL16(p[2]), y1 = SHFL16(p[3]); b0 = hi ? y0 : p[0]; b1 = hi ? y1 : p[1]; }
      else if (kc == 1) { float y0 = SHFL16(p[6]), y1 = SHFL16(p[7]); b0 = hi ? y0 : p[4]; b1 = hi ? y1 : p[5]; }
      else if (kc == 2) { float y0 = SHFL16(p[0]), y1 = SHFL16(p[1]); b0 = hi ? p[2] : y0; b1 = hi ? p[3] : y1; }
      else              { float y0 = SHFL16(p[4]), y1 = SHFL16(p[5]); b0 = hi ? p[6] : y0; b1 = hi ? p[7] : y1; }
      v2f pb; pb[0] = b0; pb[1] = b1;
      const float* vr = vbase + (4 * kc + 2 * hi) * HDIM;
#pragma unroll
      for (int mt = 0; mt < 4; ++mt) {
        v2f vf; vf[0] = vr[16 * mt]; vf[1] = vr[HDIM + 16 * mt];
        o[mt] = wmma4(vf, pb, o[mt]);
      }
    }
  }

  // epilogue: ctx^T = O^T / l ; out^T = W_O x ctx^T
  const float linv = 1.0f / lrow;
#pragma unroll
  for (int mt = 0; mt < 4; ++mt)
#pragma unroll
    for (int r = 0; r < 8; ++r) o[mt][r] *= linv;

  v8f oa[4] = {};
#pragma unroll
  for (int c = 0; c < 16; ++c) {                  // d = 64 in chunks of 4
    const int mt = c >> 2;
    const int kc = c & 3;
    float b0, b1;
    if (kc == 0) { float y0 = SHFL16(o[mt][2]), y1 = SHFL16(o[mt][3]); b0 = hi ? y0 : o[mt][0]; b1 = hi ? y1 : o[mt][1]; }
    else if (kc == 1) { float y0 = SHFL16(o[mt][6]), y1 = SHFL16(o[mt][7]); b0 = hi ? y0 : o[mt][4]; b1 = hi ? y1 : o[mt][5]; }
    else if (kc == 2) { float y0 = SHFL16(o[mt][0]), y1 = SHFL16(o[mt][1]); b0 = hi ? o[mt][2] : y0; b1 = hi ? o[mt][3] : y1; }
    else              { float y0 = SHFL16(o[mt][4]), y1 = SHFL16(o[mt][5]); b0 = hi ? o[mt][6] : y0; b1 = hi ? o[mt][7] : y1; }
    v2f bf; bf[0] = b0; bf[1] = b1;
    const float* wr = W_O + (size_t)n * HDIM + 4 * c + 2 * hi;
#pragma unroll
    for (int ot = 0; ot < 4; ++ot) {
      v2f wf = *(const v2f*)(wr + 16 * ot * HDIM);
      oa[ot] = wmma4(wf, bf, oa[ot]);
    }
  }

  // out^T C-layout: row o-dim = 16*ot + r + 8*hi, col query = n
  const int b = bh >> 4;
  const int h = bh & 15;
  float* orow = out + ((size_t)b * T_LEN + q0 + n) * 1024 + h * HDIM + 8 * hi;
#pragma unroll
  for (int ot = 0; ot < 4; ++ot)
#pragma unroll
    for (int r = 0; r < 8; ++r)
      orow[16 * ot + r] = oa[ot][r];
}

// ---------------------------------------------------------------------------
extern "C" void kernel_launch(void* const* d_in, const int* in_sizes, int n_in,
                              void* d_out, int out_size, void* d_ws, size_t ws_size,
                              hipStream_t stream) {
  (void)in_sizes; (void)n_in; (void)out_size; (void)ws_size;
  const float* x     = (const float*)d_in[0];
  const float* W_DQ  = (const float*)d_in[1];
  const float* W_UQ  = (const float*)d_in[2];
  const float* W_DKV = (const float*)d_in[3];
  const float* W_UKV = (const float*)d_in[4];
  const float* W_QR  = (const float*)d_in[5];
  const float* W_KR  = (const float*)d_in[6];
  const float* W_O   = (const float*)d_in[7];
  const float* qnw   = (const float*)d_in[8];
  const float* knw   = (const float*)d_in[9];

  float* ws  = (float*)d_ws;
  float* Wq  = ws;                                  // 4096 floats
  float* Wkv = ws + 4096;                           // 8192 floats
  float* Qb  = ws + 12288;                          // 32*2048*128
  float* Kb  = Qb + (size_t)32 * T_LEN * 128;       // 32*2048*128
  float* Vb  = Kb + (size_t)32 * T_LEN * 128;       // 32*2048*64

  combine_w_kernel<<<48, 256, 0, stream>>>(W_UQ, W_DQ, W_UKV, W_DKV, Wq, Wkv);
  mla_qkv_kernel<<<256, 256, 0, stream>>>(x, Wq, Wkv, W_QR, W_KR, qnw, knw,
                                          Qb, Kb, Vb);
  dim3 g(32, 32);
  mla_attn_kernel<<<g, 128, 0, stream>>>(Qb, Kb, Vb, W_O, (float*)d_out);
}